// WHOffSmoothL1Loss_26766236189165
// MI455X (gfx1250) — compile-verified
//
#include <hip/hip_runtime.h>
#include <math.h>

// ---- problem constants (from reference: B=32, C=10, H=W=256, K=500) ----
#define BB     32
#define CC     10
#define HW     65536          // 256*256
#define KK     500
#define NBOX   (BB * KK)      // 16000 boxes
#define THREADS 256
#define NBLK   ((NBOX + THREADS - 1) / THREADS)   // 63 blocks

typedef __attribute__((ext_vector_type(2))) float v2f;
typedef __attribute__((ext_vector_type(8))) float v8f;

// Kernel 1: per-box CIoU-style penalty term, then wave32 WMMA reduction.
__global__ __launch_bounds__(THREADS) void ciou_partial_kernel(
    const float* __restrict__ output,   // [B, C, H*W]
    const float* __restrict__ mask,     // [B, K]
    const int*   __restrict__ ind,      // [B, K]
    const float* __restrict__ target,   // [B, K, C]
    float* __restrict__ partial)        // [NBLK]
{
    __shared__ float wsum[THREADS / 32];

    const int t = blockIdx.x * blockDim.x + threadIdx.x;
    float val = 0.0f;

    if (t < NBOX) {
        const int b   = t / KK;
        const int idx = ind[t];

        // gather pred[b, t%KK, c] = output[b, c, idx]; stride HW between channels
        const float* ap = output + (size_t)b * CC * HW + (size_t)idx;
        const float a0 = ap[0 * HW], a1 = ap[1 * HW], a2 = ap[2 * HW], a3 = ap[3 * HW];
        const float a4 = ap[4 * HW], a5 = ap[5 * HW], a6 = ap[6 * HW], a7 = ap[7 * HW];

        const float* bp = target + (size_t)t * CC;
        const float b0 = bp[0], b1 = bp[1], b2 = bp[2], b3 = bp[3];
        const float b4 = bp[4], b5 = bp[5], b6 = bp[6], b7 = bp[7];

        const float d26 = a2 - a6, d37 = a3 - a7;
        const float w   = d26 * d26 + d37 * d37;
        const float d04 = a0 - a4, d15 = a1 - a5;
        const float h   = d04 * d04 + d15 * d15;
        // NOTE: faithful to reference bug: uses a7 (not b7) in wt's second term
        const float e26 = b2 - b6, e3a7 = b3 - a7;
        const float wt  = e26 * e26 + e3a7 * e3a7;
        const float e04 = b0 - b4, e15 = b1 - b5;
        const float ht  = e04 * e04 + e15 * e15;

        const float th  = d37 / d26;
        const float tth = (b3 - b7) / e26;

        const float t1 = atanf(wt / ht) - atanf(w / h);
        const float t2 = atanf(th) - atanf(tth);
        const float four_over_pi2 = 0.40528473456935108577f; // 4/pi^2
        const float v = four_over_pi2 * (t1 * t1 + t2 * t2);

        val = v * mask[t];
    }
    // EXEC reconverges to all-1s here (required by WMMA).

    // ---- wave32 sum via V_WMMA_F32_16X16X4_F32 with B = ones ----
    // A-matrix layout (16x4 f32): lane L<16 holds A[L][0],A[L][1];
    // lane L>=16 holds A[L-16][2],A[L-16][3].  With a.y = 0 and B all-ones:
    //   D[m][n] = val[lane m] + val[lane m+16]   for every n.
    v2f a;  a.x = val;  a.y = 0.0f;
    v2f bm; bm.x = 1.0f; bm.y = 1.0f;
    v8f c = {};
    c = __builtin_amdgcn_wmma_f32_16x16x4_f32(
            /*neg_a=*/false, a, /*neg_b=*/false, bm,
            /*c_mod=*/(short)0, c, /*reuse_a=*/false, /*reuse_b=*/false);

    // D VGPR j: lanes 0-15 hold row j, lanes 16-31 hold row j+8.
    // Per-lane sum over the 8 D VGPRs -> lane n has rows 0..7, lane n+16 rows 8..15.
    float s = c[0] + c[1] + c[2] + c[3] + c[4] + c[5] + c[6] + c[7];

    // Combine the two half-wave partials: ds_swizzle SWAPX16
    // (group-of-32 mode: xor_mask=0x10, or_mask=0, and_mask=0x1f -> 0x401F)
    int swz = __builtin_amdgcn_ds_swizzle(__float_as_int(s), 0x401F);
    float wavesum = s + __int_as_float(swz);

    const int lane = threadIdx.x & 31;
    const int wid  = threadIdx.x >> 5;
    if (lane == 0) wsum[wid] = wavesum;
    __syncthreads();

    if (threadIdx.x == 0) {
        float bs = 0.0f;
#pragma unroll
        for (int i = 0; i < THREADS / 32; ++i) bs += wsum[i];
        partial[blockIdx.x] = bs;
    }
}

// Kernel 2: deterministic fixed-order final sum (63 partials -> scalar).
// Overwrites d_out, so no pre-zeroing needed across graph replays.
__global__ void final_reduce_kernel(const float* __restrict__ partial,
                                    float* __restrict__ out, int n)
{
    if (threadIdx.x == 0 && blockIdx.x == 0) {
        float s = 0.0f;
        for (int i = 0; i < n; ++i) s += partial[i];
        out[0] = s;
    }
}

extern "C" void kernel_launch(void* const* d_in, const int* in_sizes, int n_in,
                              void* d_out, int out_size, void* d_ws, size_t ws_size,
                              hipStream_t stream)
{
    const float* output = (const float*)d_in[0];  // [32,10,256,256] f32
    const float* mask   = (const float*)d_in[1];  // [32,500] f32
    const int*   ind    = (const int*)d_in[2];    // [32,500] int
    const float* target = (const float*)d_in[3];  // [32,500,10] f32

    float* partial = (float*)d_ws;                // NBLK floats of scratch

    ciou_partial_kernel<<<NBLK, THREADS, 0, stream>>>(output, mask, ind, target, partial);
    final_reduce_kernel<<<1, 32, 0, stream>>>(partial, (float*)d_out, NBLK);
}